// IBGConv_76106820485776
// MI455X (gfx1250) — compile-verified
//
#include <hip/hip_runtime.h>
#include <math.h>

typedef float v2f __attribute__((ext_vector_type(2)));
typedef float v4f __attribute__((ext_vector_type(4)));
typedef float v8f __attribute__((ext_vector_type(8)));

#define N_NODES 50000
#define N_EDGES 1600000
#define G_NUM   64
#define C_DIM   2

// ---------------- degree / normalization precompute ----------------

__global__ void k_init_deg_cnt(float* __restrict__ deg, float* __restrict__ cnt) {
  int v = blockIdx.x * blockDim.x + threadIdx.x;
  if (v < N_NODES) { deg[v] = 1.0f; cnt[v] = 1.0f; }  // self-loop contributes 1
}

__global__ void k_edge_deg(const int* __restrict__ col, const float* __restrict__ ea,
                           float* __restrict__ deg, float* __restrict__ cnt) {
  int e = blockIdx.x * blockDim.x + threadIdx.x;
  if (e < N_EDGES) {
    int c = col[e];
    atomicAdd(&deg[c], fabsf(ea[e]));
    atomicAdd(&cnt[c], 1.0f);
  }
}

__global__ void k_dis_sew(const float* __restrict__ deg, float* __restrict__ dis,
                          float* __restrict__ sew) {
  int v = blockIdx.x * blockDim.x + threadIdx.x;
  if (v < N_NODES) {
    float d = deg[v];
    float r = (d > 0.0f) ? rsqrtf(d) : 0.0f;
    dis[v] = r;
    sew[v] = r * r;   // self-loop edge weight dis[v]*1*dis[v]
  }
}

__global__ void k_edge_sew(const int* __restrict__ row, const int* __restrict__ col,
                           const float* __restrict__ ea, const float* __restrict__ dis,
                           float* __restrict__ sew) {
  int e = blockIdx.x * blockDim.x + threadIdx.x;
  if (e < N_EDGES) {
    int r = row[e], c = col[e];
    atomicAdd(&sew[c], dis[r] * fabsf(ea[e]) * dis[c]);
  }
}

// ---------------- WMMA f32 GEMM: Out[M,NOUT] = A[M,K] @ B[K,NOUT] ----------------
// One wave per 16x16 output tile. M == N_NODES (multiple of 16).
// A frag (16x4 f32): lanes 0-15 hold M=lane%16, K = {0,1}; lanes 16-31 K = {2,3}.
// B frag (4x16 f32): lanes 0-15 hold N=lane%16, K = {0,1}; lanes 16-31 K = {2,3}.
// D (16x16 f32): VGPR r -> M = r + 8*(lane/16), N = lane%16.

template <int K, int NOUT>
__global__ void k_gemm_wmma(const float* __restrict__ A, const float* __restrict__ B,
                            float* __restrict__ Out) {
  constexpr int  TILES_N = (NOUT + 15) / 16;
  constexpr bool FULL_N  = (NOUT % 16) == 0;   // no column guards needed
  constexpr bool FULL_K  = (K % 4) == 0;       // no K guards needed

  const int tiles_total = (N_NODES / 16) * TILES_N;
  int wave = blockIdx.x * (blockDim.x >> 5) + (threadIdx.x >> 5);
  if (wave >= tiles_total) return;  // wave-uniform: EXEC all-ones at every WMMA
  int tm = wave / TILES_N;
  int tn = wave - tm * TILES_N;
  int lane = threadIdx.x & 31;
  int hl = lane >> 4;               // K-pair selector
  int lr = lane & 15;
  int m = tm * 16 + lr;
  int n = tn * 16 + lr;
  bool nok = FULL_N || (n < NOUT);

  v8f acc = {0.f, 0.f, 0.f, 0.f, 0.f, 0.f, 0.f, 0.f};
#pragma unroll
  for (int k0 = 0; k0 < K; k0 += 4) {
    int ka = k0 + 2 * hl;
    v2f a, b;
    if constexpr (FULL_K) {
      a = *(const v2f*)(A + m * K + ka);        // global_load_b64, 8B aligned
    } else {
      a.x = (ka < K)     ? A[m * K + ka]     : 0.0f;
      a.y = (ka + 1 < K) ? A[m * K + ka + 1] : 0.0f;
    }
    if constexpr (FULL_N && FULL_K) {
      b.x = B[ka * NOUT + n];
      b.y = B[(ka + 1) * NOUT + n];
    } else {
      b.x = (nok && ka < K)     ? B[ka * NOUT + n]       : 0.0f;
      b.y = (nok && ka + 1 < K) ? B[(ka + 1) * NOUT + n] : 0.0f;
    }
    acc = __builtin_amdgcn_wmma_f32_16x16x4_f32(false, a, false, b,
                                                (short)0, acc, false, false);
  }
#pragma unroll
  for (int r = 0; r < 8; ++r) {
    int orow = tm * 16 + r + 8 * hl;
    int ocol = tn * 16 + lr;
    if (FULL_N || ocol < NOUT) Out[orow * NOUT + ocol] = acc[r];
  }
}

// ---------------- pack linW -> [A | B] of shape [dd, 2*dd] ----------------

__global__ void k_pack_wcat(const float* __restrict__ linW, float* __restrict__ wcat, int dd) {
  int idx = blockIdx.x * blockDim.x + threadIdx.x;
  int total = dd * 2 * dd;
  if (idx < total) {
    int k = idx / (2 * dd);
    int j = idx - k * (2 * dd);
    wcat[idx] = (j < dd) ? linW[k * dd + j] : linW[(dd + k) * dd + (j - dd)];
  }
}

// ---------------- aggregation of m = h@B over edges ----------------
// qm layout: [N, 2*dd], columns [0,dd) = q = h@A, [dd,2dd) = m = h@B

template <int DD>
__global__ void k_aggm_init(const float* __restrict__ qm, float* __restrict__ aggm) {
  int idx = blockIdx.x * blockDim.x + threadIdx.x;
  if (idx < N_NODES * DD) {
    int v = idx / DD;
    int j = idx - v * DD;
    aggm[idx] = qm[v * (2 * DD) + DD + j];   // self-loop: m[v]
  }
}

// DD=64: one thread per (edge, 4 features) — b128 gather, 4 atomics
__global__ void k_edge_agg64(const int* __restrict__ row, const int* __restrict__ col,
                             const float* __restrict__ qm, float* __restrict__ aggm) {
  int gid = blockIdx.x * blockDim.x + threadIdx.x;   // N_EDGES*16 = 25.6M
  if (gid < N_EDGES * 16) {
    int e  = gid >> 4;
    int j4 = (gid & 15) << 2;
    v4f mv = *(const v4f*)(qm + (size_t)row[e] * 128 + 64 + j4);  // 16B aligned
    float* dst = aggm + (size_t)col[e] * 64 + j4;
    atomicAdd(dst + 0, mv.x);
    atomicAdd(dst + 1, mv.y);
    atomicAdd(dst + 2, mv.z);
    atomicAdd(dst + 3, mv.w);
  }
}

// DD=2: one thread per edge — b64 gather, 2 atomics
__global__ void k_edge_agg2(const int* __restrict__ row, const int* __restrict__ col,
                            const float* __restrict__ qm, float* __restrict__ aggm) {
  int e = blockIdx.x * blockDim.x + threadIdx.x;
  if (e < N_EDGES) {
    v2f mv = *(const v2f*)(qm + (size_t)row[e] * 4 + 2);   // 8B aligned
    float* dst = aggm + (size_t)col[e] * 2;
    atomicAdd(dst + 0, mv.x);
    atomicAdd(dst + 1, mv.y);
  }
}

template <int DD>
__global__ void k_combine(const float* __restrict__ qm, const float* __restrict__ aggm,
                          const float* __restrict__ cnt, const float* __restrict__ sew,
                          const float* __restrict__ linb, const float* __restrict__ wvec,
                          const float* __restrict__ bb, float* __restrict__ z, int relu) {
  int idx = blockIdx.x * blockDim.x + threadIdx.x;
  if (idx < N_NODES * DD) {
    int v = idx / DD;
    int j = idx - v * DD;
    float val = cnt[v] * (qm[v * (2 * DD) + j] + linb[j]) + aggm[idx]
              + sew[v] * wvec[j] + bb[j];
    if (relu) val = fmaxf(val, 0.0f);
    z[idx] = val;
  }
}

// ---------------- mean pool + log_softmax ----------------

__global__ void k_pool_init(float* __restrict__ gsum, float* __restrict__ gcnt) {
  int i = threadIdx.x;
  if (i < G_NUM * C_DIM) gsum[i] = 0.0f;
  if (i < G_NUM) gcnt[i] = 0.0f;
}

__global__ void k_pool(const float* __restrict__ z, const int* __restrict__ batch,
                       float* __restrict__ gsum, float* __restrict__ gcnt) {
  int v = blockIdx.x * blockDim.x + threadIdx.x;
  if (v < N_NODES) {
    int g = batch[v];
    atomicAdd(&gsum[g * C_DIM + 0], z[v * C_DIM + 0]);
    atomicAdd(&gsum[g * C_DIM + 1], z[v * C_DIM + 1]);
    atomicAdd(&gcnt[g], 1.0f);
  }
}

__global__ void k_logsoftmax(const float* __restrict__ gsum, const float* __restrict__ gcnt,
                             float* __restrict__ out) {
  int g = threadIdx.x;
  if (g < G_NUM) {
    float c  = fmaxf(gcnt[g], 1.0f);
    float a  = gsum[g * 2 + 0] / c;
    float b  = gsum[g * 2 + 1] / c;
    float mx = fmaxf(a, b);
    float lse = mx + logf(expf(a - mx) + expf(b - mx));
    out[g * 2 + 0] = a - lse;
    out[g * 2 + 1] = b - lse;
  }
}

// ---------------- host orchestration ----------------

static inline int cdiv_ll(long long a, long long b) { return (int)((a + b - 1) / b); }

extern "C" void kernel_launch(void* const* d_in, const int* in_sizes, int n_in,
                              void* d_out, int out_size, void* d_ws, size_t ws_size,
                              hipStream_t stream) {
  (void)in_sizes; (void)n_in; (void)out_size; (void)ws_size;

  const float* x     = (const float*)d_in[0];
  const int*   eidx  = (const int*)d_in[1];
  const int*   row   = eidx;             // edge_index[0] -> source (x_j)
  const int*   col   = eidx + N_EDGES;   // edge_index[1] -> target / agg index
  const float* ea    = (const float*)d_in[2];
  const int*   batch = (const int*)d_in[4];

  float* ws = (float*)d_ws;
  size_t o = 0;
  float* deg  = ws + o; o += N_NODES;
  float* cnt  = ws + o; o += N_NODES;
  float* dis  = ws + o; o += N_NODES;
  float* sew  = ws + o; o += N_NODES;
  float* h    = ws + o; o += (size_t)N_NODES * 64;
  float* qm   = ws + o; o += (size_t)N_NODES * 128;
  float* aggm = ws + o; o += (size_t)N_NODES * 64;
  float* zA   = ws + o; o += (size_t)N_NODES * 64;
  float* gsum = ws + o; o += G_NUM * C_DIM;
  float* gcnt = ws + o; o += G_NUM;
  float* wcat = ws + o; o += 64 * 128;

  const int TB = 256;
  const int WPB = TB / 32;   // waves per block
  const int TM  = N_NODES / 16;

  k_init_deg_cnt<<<cdiv_ll(N_NODES, TB), TB, 0, stream>>>(deg, cnt);
  k_edge_deg   <<<cdiv_ll(N_EDGES, TB), TB, 0, stream>>>(col, ea, deg, cnt);
  k_dis_sew    <<<cdiv_ll(N_NODES, TB), TB, 0, stream>>>(deg, dis, sew);
  k_edge_sew   <<<cdiv_ll(N_EDGES, TB), TB, 0, stream>>>(row, col, ea, dis, sew);

  for (int l = 0; l < 3; ++l) {
    const float* W    = (const float*)d_in[5 + 4 * l + 0];
    const float* linW = (const float*)d_in[5 + 4 * l + 1];
    const float* linb = (const float*)d_in[5 + 4 * l + 2];
    const float* bb   = (const float*)d_in[5 + 4 * l + 3];
    const int dd = (l == 2) ? 2 : 64;
    const float* zin = (l == 0) ? x : zA;

    // GEMM1: h = zin @ W   [N, dd]
    if (l == 0)      k_gemm_wmma<128, 64><<<cdiv_ll((long long)TM * 4, WPB), TB, 0, stream>>>(zin, W, h);
    else if (l == 1) k_gemm_wmma< 64, 64><<<cdiv_ll((long long)TM * 4, WPB), TB, 0, stream>>>(zin, W, h);
    else             k_gemm_wmma< 64,  2><<<cdiv_ll((long long)TM * 1, WPB), TB, 0, stream>>>(zin, W, h);

    // wcat = [A | B]  shape [dd, 2*dd]
    k_pack_wcat<<<cdiv_ll(dd * 2 * dd, TB), TB, 0, stream>>>(linW, wcat, dd);

    // GEMM2: qm = h @ wcat   [N, 2*dd]
    if (dd == 64) k_gemm_wmma<64, 128><<<cdiv_ll((long long)TM * 8, WPB), TB, 0, stream>>>(h, wcat, qm);
    else          k_gemm_wmma< 2,   4><<<cdiv_ll((long long)TM * 1, WPB), TB, 0, stream>>>(h, wcat, qm);

    const float* wvec = linW + (size_t)(2 * dd) * dd;  // last row of linW
    if (dd == 64) {
      k_aggm_init<64><<<cdiv_ll((long long)N_NODES * 64, TB), TB, 0, stream>>>(qm, aggm);
      k_edge_agg64   <<<cdiv_ll((long long)N_EDGES * 16, TB), TB, 0, stream>>>(row, col, qm, aggm);
      k_combine  <64><<<cdiv_ll((long long)N_NODES * 64, TB), TB, 0, stream>>>(
          qm, aggm, cnt, sew, linb, wvec, bb, zA, (l < 2) ? 1 : 0);
    } else {
      k_aggm_init<2><<<cdiv_ll((long long)N_NODES * 2, TB), TB, 0, stream>>>(qm, aggm);
      k_edge_agg2   <<<cdiv_ll((long long)N_EDGES, TB), TB, 0, stream>>>(row, col, qm, aggm);
      k_combine  <2><<<cdiv_ll((long long)N_NODES * 2, TB), TB, 0, stream>>>(
          qm, aggm, cnt, sew, linb, wvec, bb, zA, 0);
    }
  }

  k_pool_init<<<1, 256, 0, stream>>>(gsum, gcnt);
  k_pool     <<<cdiv_ll(N_NODES, TB), TB, 0, stream>>>(zA, batch, gsum, gcnt);
  k_logsoftmax<<<1, 64, 0, stream>>>(gsum, gcnt, (float*)d_out);
}